// DCRB_14121852470057
// MI455X (gfx1250) — compile-verified
//
#include <hip/hip_runtime.h>
#include <hip/hip_bf16.h>

typedef float v2f __attribute__((ext_vector_type(2)));
typedef float v8f __attribute__((ext_vector_type(8)));

#define C0 128
#define C1 32
#define HH 96
#define WW 96
#define HW (HH * WW)          // 9216
#define NB 4                  // batch
#define TS 40                 // LDS row stride (floats): 2*40 mod 64 = 16 -> disjoint bank halves

__device__ __forceinline__ v8f wmma4(v2f a, v2f b, v8f c) {
  // D = A(16x4,f32) * B(4x16,f32) + C(16x16,f32)
  return __builtin_amdgcn_wmma_f32_16x16x4_f32(
      /*neg_a=*/false, a, /*neg_b=*/false, b,
      /*c_mod=*/(short)0, c, /*reuse_a=*/false, /*reuse_b=*/false);
}

__device__ __forceinline__ v8f vzero8() {
  v8f z = {0.f, 0.f, 0.f, 0.f, 0.f, 0.f, 0.f, 0.f};
  return z;
}

// ---------------------------------------------------------------------------
// Kernel 1: 3x3 conv, C0=128 -> 18 offset channels, SAME padding.
// One thread per pixel, all 18 outputs in registers; weights chunk-staged in LDS.
// grid = B*HW/256 = 144 blocks (exact), block = 256.
// ---------------------------------------------------------------------------
__global__ __launch_bounds__(256) void off_conv_kernel(
    const float* __restrict__ x0, const float* __restrict__ w,
    const float* __restrict__ bias, float* __restrict__ off) {
  __shared__ float sw[18 * 32 * 9];  // 20.7 KB chunk: 32 input channels at a time
  int p  = blockIdx.x * 256 + threadIdx.x;     // global pixel over B*HW
  int b  = p / HW;
  int hw = p - b * HW;
  int h  = hw / WW;
  int x  = hw - h * WW;

  float acc[18];
#pragma unroll
  for (int o = 0; o < 18; ++o) acc[o] = bias[o];

  const float* X = x0 + (size_t)b * C0 * HW;
  for (int cb = 0; cb < C0; cb += 32) {
    __syncthreads();
    for (int i = threadIdx.x; i < 18 * 32 * 9; i += 256) {
      int o = i / (32 * 9);
      int r = i - o * (32 * 9);
      int c = r / 9;
      int t = r - c * 9;
      sw[i] = w[(o * C0 + cb + c) * 9 + t];
    }
    __syncthreads();
    for (int c = 0; c < 32; ++c) {
      const float* Xc = X + (size_t)(cb + c) * HW;
#pragma unroll
      for (int t = 0; t < 9; ++t) {
        int yy = h + t / 3 - 1;
        int xx = x + t % 3 - 1;
        float xv = (yy >= 0 && yy < HH && xx >= 0 && xx < WW) ? Xc[yy * WW + xx] : 0.f;
#pragma unroll
        for (int o = 0; o < 18; ++o)
          acc[o] = fmaf(sw[(o * 32 + c) * 9 + t], xv, acc[o]);
      }
    }
  }
  float* O = off + (size_t)b * 18 * HW + hw;
#pragma unroll
  for (int o = 0; o < 18; ++o) O[(size_t)o * HW] = acc[o];
}

// ---------------------------------------------------------------------------
// Kernel 2: fused SFT.  out = xin * (scale+1) + shift,
//   scale = W1s @ lrelu(W0s @ x1 + b0s) + b1s  (shift chain analogous).
// Block = 256 (8 waves). Block tile: all 128 output rows x 32 pixels.
// Stage-1 result (128x32, both chains) lives in LDS; both stages use
// V_WMMA_F32_16X16X4_F32. grid = B*HW/32 = 1152 blocks.
// ---------------------------------------------------------------------------
__global__ __launch_bounds__(256) void sft_kernel(
    const float* __restrict__ x1, const float* __restrict__ xin,
    const float* __restrict__ w_sc0, const float* __restrict__ b_sc0,
    const float* __restrict__ w_sc1, const float* __restrict__ b_sc1,
    const float* __restrict__ w_sh0, const float* __restrict__ b_sh0,
    const float* __restrict__ w_sh1, const float* __restrict__ b_sh1,
    float* __restrict__ out) {
  __shared__ float t_sc[C0 * TS];  // 20 KB
  __shared__ float t_sh[C0 * TS];  // 20 KB

  const int tilesPerBatch = HW / 32;  // 288
  int b  = blockIdx.x / tilesPerBatch;
  int p0 = (blockIdx.x - b * tilesPerBatch) * 32;

  int wave = threadIdx.x >> 5;
  int lane = threadIdx.x & 31;
  int lrow = lane & 15;    // fragment row/col within 16
  int lhi  = lane >> 4;    // 0 or 1 (K/M half select)
  int m0   = wave * 16;

  const float* X1 = x1 + (size_t)b * C1 * HW + p0;

  // ---- stage 1: t = lrelu(W0 @ X1tile + b0), both chains ----
#pragma unroll
  for (int nt = 0; nt < 2; ++nt) {
    int n0 = nt * 16;
    v8f asc = vzero8(), ash = vzero8();
#pragma unroll
    for (int kk = 0; kk < C1; kk += 4) {
      int krow = kk + 2 * lhi;
      v2f bf;
      bf.x = X1[(size_t)krow * HW + n0 + lrow];
      bf.y = X1[(size_t)(krow + 1) * HW + n0 + lrow];
      v2f a0 = *(const v2f*)(w_sc0 + (m0 + lrow) * C1 + krow);
      v2f a1 = *(const v2f*)(w_sh0 + (m0 + lrow) * C1 + krow);
      asc = wmma4(a0, bf, asc);
      ash = wmma4(a1, bf, ash);
    }
#pragma unroll
    for (int r = 0; r < 8; ++r) {
      int M = m0 + r + 8 * lhi;
      float vsc = asc[r] + b_sc0[M];
      float vsh = ash[r] + b_sh0[M];
      vsc = vsc > 0.f ? vsc : 0.1f * vsc;   // leaky_relu 0.1
      vsh = vsh > 0.f ? vsh : 0.1f * vsh;
      t_sc[M * TS + n0 + lrow] = vsc;
      t_sh[M * TS + n0 + lrow] = vsh;
    }
  }
  __syncthreads();

  // ---- stage 2 + elementwise epilogue ----
  const float* XIN = xin + (size_t)b * C0 * HW + p0;
  float* OUT       = out + (size_t)b * C0 * HW + p0;
#pragma unroll
  for (int nt = 0; nt < 2; ++nt) {
    int n0 = nt * 16;
    v8f asc = vzero8(), ash = vzero8();
#pragma unroll 4
    for (int kk = 0; kk < C0; kk += 4) {
      int krow = kk + 2 * lhi;
      v2f bsc, bsh;
      bsc.x = t_sc[krow * TS + n0 + lrow];
      bsc.y = t_sc[(krow + 1) * TS + n0 + lrow];
      bsh.x = t_sh[krow * TS + n0 + lrow];
      bsh.y = t_sh[(krow + 1) * TS + n0 + lrow];
      v2f a0 = *(const v2f*)(w_sc1 + (m0 + lrow) * C0 + krow);
      v2f a1 = *(const v2f*)(w_sh1 + (m0 + lrow) * C0 + krow);
      asc = wmma4(a0, bsc, asc);
      ash = wmma4(a1, bsh, ash);
    }
#pragma unroll
    for (int r = 0; r < 8; ++r) {
      int M = m0 + r + 8 * lhi;
      float scale = asc[r] + b_sc1[M];
      float shift = ash[r] + b_sh1[M];
      float xv = XIN[(size_t)M * HW + n0 + lrow];
      OUT[(size_t)M * HW + n0 + lrow] = xv * (scale + 1.0f) + shift;
    }
  }
}

// ---------------------------------------------------------------------------
// Kernel 3: deformable conv 3x3 (stride 1, pad 1) as 9 accumulated WMMA GEMMs.
// Per 32-pixel tile: for each tap, build bilinear-sampled tile samp[128x32] in
// LDS, then acc += W_tap[128x128] @ samp. Accumulators persist in VGPRs.
// grid = B*HW/32 = 1152 blocks, block = 256 (8 waves).
// ---------------------------------------------------------------------------
__device__ __forceinline__ float bsample(const float* __restrict__ Fc, int y, int x) {
  bool v = (y >= 0) & (y < HH) & (x >= 0) & (x < WW);
  int yc = min(max(y, 0), HH - 1);
  int xc = min(max(x, 0), WW - 1);
  float val = Fc[yc * WW + xc];
  return v ? val : 0.f;
}

__global__ __launch_bounds__(256) void dcn_kernel(
    const float* __restrict__ fea,   // [B,C0,HW] sampling source
    const float* __restrict__ off,   // [B,18,HW]
    const float* __restrict__ wd,    // [C0,C0,9]
    const float* __restrict__ bd,    // [C0]
    const float* __restrict__ resid, // [B,C0,HW] or nullptr
    float* __restrict__ out,
    int do_relu) {
  __shared__ float samp[C0 * TS];  // 20 KB sampled tile
  __shared__ float s_wy[32], s_wx[32];
  __shared__ int   s_iy[32], s_ix[32];

  const int tilesPerBatch = HW / 32;
  int b  = blockIdx.x / tilesPerBatch;
  int p0 = (blockIdx.x - b * tilesPerBatch) * 32;  // tile lies inside one image row

  int wave = threadIdx.x >> 5;
  int lane = threadIdx.x & 31;
  int lrow = lane & 15;
  int lhi  = lane >> 4;
  int m0   = wave * 16;

  v8f acc0 = vzero8();  // n-tile 0 (pixels p0..p0+15)
  v8f acc1 = vzero8();  // n-tile 1 (pixels p0+16..p0+31)

  const float* F   = fea + (size_t)b * C0 * HW;
  const float* OFF = off + (size_t)b * 18 * HW;

  for (int tap = 0; tap < 9; ++tap) {
    __syncthreads();  // protect samp WAR vs previous tap's GEMM reads
    if (threadIdx.x < 32) {
      int i = threadIdx.x;
      int p = p0 + i;
      int h = p / WW;
      int x = p - h * WW;
      float dy = OFF[(size_t)(2 * tap) * HW + p];
      float dx = OFF[(size_t)(2 * tap + 1) * HW + p];
      float py = dy + (float)h + (float)(tap / 3 - 1);
      float px = dx + (float)x + (float)(tap % 3 - 1);
      float fy = floorf(py), fx = floorf(px);
      s_wy[i] = py - fy;
      s_wx[i] = px - fx;
      s_iy[i] = (int)fy;
      s_ix[i] = (int)fx;
    }
    __syncthreads();

    // fill sampled tile: 128 channels x 32 pixels
    for (int t = threadIdx.x; t < C0 * 32; t += 256) {
      int i = t & 31;
      int c = t >> 5;
      int iy = s_iy[i], ix = s_ix[i];
      float wy = s_wy[i], wx = s_wx[i];
      const float* Fc = F + (size_t)c * HW;
      float v00 = bsample(Fc, iy, ix);
      float v01 = bsample(Fc, iy, ix + 1);
      float v10 = bsample(Fc, iy + 1, ix);
      float v11 = bsample(Fc, iy + 1, ix + 1);
      samp[c * TS + i] = v00 * (1.f - wy) * (1.f - wx) + v01 * (1.f - wy) * wx +
                         v10 * wy * (1.f - wx) + v11 * wy * wx;
    }
    __syncthreads();

    // acc += W_tap[m0:m0+16, :] @ samp
#pragma unroll 4
    for (int kk = 0; kk < C0; kk += 4) {
      int krow = kk + 2 * lhi;
      v2f b0, b1, a;
      b0.x = samp[krow * TS + lrow];
      b0.y = samp[(krow + 1) * TS + lrow];
      b1.x = samp[krow * TS + 16 + lrow];
      b1.y = samp[(krow + 1) * TS + 16 + lrow];
      a.x = wd[((size_t)(m0 + lrow) * C0 + krow) * 9 + tap];
      a.y = wd[((size_t)(m0 + lrow) * C0 + krow + 1) * 9 + tap];
      acc0 = wmma4(a, b0, acc0);
      acc1 = wmma4(a, b1, acc1);
    }
  }

  // epilogue: bias, optional relu, optional residual
  const float* R = resid ? (resid + (size_t)b * C0 * HW + p0) : nullptr;
  float* O       = out + (size_t)b * C0 * HW + p0;
#pragma unroll
  for (int r = 0; r < 8; ++r) {
    int M = m0 + r + 8 * lhi;
    float v0 = acc0[r] + bd[M];
    float v1 = acc1[r] + bd[M];
    if (do_relu) {
      v0 = fmaxf(v0, 0.f);
      v1 = fmaxf(v1, 0.f);
    }
    if (R) {
      v0 += R[(size_t)M * HW + lrow];
      v1 += R[(size_t)M * HW + 16 + lrow];
    }
    O[(size_t)M * HW + lrow] = v0;
    O[(size_t)M * HW + 16 + lrow] = v1;
  }
}

// ---------------------------------------------------------------------------
// Host orchestration
// ---------------------------------------------------------------------------
extern "C" void kernel_launch(void* const* d_in, const int* in_sizes, int n_in,
                              void* d_out, int out_size, void* d_ws, size_t ws_size,
                              hipStream_t stream) {
  (void)in_sizes; (void)n_in; (void)out_size; (void)ws_size;

  const float* x0      = (const float*)d_in[0];
  const float* x1      = (const float*)d_in[1];
  const float* w_off   = (const float*)d_in[2];
  const float* b_off   = (const float*)d_in[3];
  const float* s1_sc0w = (const float*)d_in[4];
  const float* s1_sc0b = (const float*)d_in[5];
  const float* s1_sc1w = (const float*)d_in[6];
  const float* s1_sc1b = (const float*)d_in[7];
  const float* s1_sh0w = (const float*)d_in[8];
  const float* s1_sh0b = (const float*)d_in[9];
  const float* s1_sh1w = (const float*)d_in[10];
  const float* s1_sh1b = (const float*)d_in[11];
  const float* w_d1    = (const float*)d_in[12];
  const float* b_d1    = (const float*)d_in[13];
  const float* s2_sc0w = (const float*)d_in[14];
  const float* s2_sc0b = (const float*)d_in[15];
  const float* s2_sc1w = (const float*)d_in[16];
  const float* s2_sc1b = (const float*)d_in[17];
  const float* s2_sh0w = (const float*)d_in[18];
  const float* s2_sh0b = (const float*)d_in[19];
  const float* s2_sh1w = (const float*)d_in[20];
  const float* s2_sh1b = (const float*)d_in[21];
  const float* w_d2    = (const float*)d_in[22];
  const float* b_d2    = (const float*)d_in[23];
  float* out           = (float*)d_out;

  // workspace layout (all 256B-aligned sizes)
  const size_t OFF_BYTES = (size_t)NB * 18 * HW * sizeof(float);   // 2,654,208
  const size_t FEA_BYTES = (size_t)NB * C0 * HW * sizeof(float);   // 18,874,368
  char* ws     = (char*)d_ws;
  float* off   = (float*)ws;                                // offsets (shared by both DCNs)
  float* buf1  = (float*)(ws + OFF_BYTES);                  // fea after SFT1 / SFT2 (reused)
  float* buf2  = (float*)(ws + OFF_BYTES + FEA_BYTES);      // fea after relu(DCN1)

  const int blocks_pix  = (NB * HW) / 256;   // 144
  const int blocks_tile = (NB * HW) / 32;    // 1152

  // 1) offsets = conv3x3(x0)
  off_conv_kernel<<<blocks_pix, 256, 0, stream>>>(x0, w_off, b_off, off);

  // 2) fea1 = SFT(x0, x1; s1)
  sft_kernel<<<blocks_tile, 256, 0, stream>>>(x1, x0,
      s1_sc0w, s1_sc0b, s1_sc1w, s1_sc1b,
      s1_sh0w, s1_sh0b, s1_sh1w, s1_sh1b, buf1);

  // 3) fea2 = relu(DCN(fea1, off; w_d1))
  dcn_kernel<<<blocks_tile, 256, 0, stream>>>(buf1, off, w_d1, b_d1,
                                              nullptr, buf2, /*do_relu=*/1);

  // 4) fea3 = SFT(fea2, x1; s2)  (reuse buf1)
  sft_kernel<<<blocks_tile, 256, 0, stream>>>(x1, buf2,
      s2_sc0w, s2_sc0b, s2_sc1w, s2_sc1b,
      s2_sh0w, s2_sh0b, s2_sh1w, s2_sh1b, buf1);

  // 5) out = x0 + DCN(fea3, off; w_d2)
  dcn_kernel<<<blocks_tile, 256, 0, stream>>>(buf1, off, w_d2, b_d2,
                                              x0, out, /*do_relu=*/0);
}